// HashEncoding_8735963480350
// MI455X (gfx1250) — compile-verified
//
#include <hip/hip_runtime.h>
#include <hip/hip_bf16.h>

// Multiresolution hash-grid encoding (no interpolation):
//   out[p, l*2+f] = table[l][ (cx + cy*2654435761 + cz*805459861) & (2^19-1) ][f]
// with c = clamp(floor(pos*res_l), 0, res_l-1).
//
// Memory-bound gather. Tables (64 MB) fit in MI455X 192 MB L2 -> gathers use
// default (RT) temporal hint; streaming input/output use NT hints so they do
// not evict the tables from L2.

typedef float v2f __attribute__((ext_vector_type(2)));

#define HG_LEVELS      16
#define HG_TABLE_MASK  0x7FFFFu          // TABLE_SIZE = 2^19
#define HG_PTS_PER_BLK 16                // 256 threads = 16 points x 16 levels

// int(16 * (128 ** (l/15))) for l = 0..15, computed in double like the reference
__constant__ float c_res[HG_LEVELS] = {
    16.0f,   22.0f,   30.0f,   42.0f,   58.0f,   80.0f,   111.0f,  153.0f,
    212.0f,  294.0f,  406.0f,  561.0f,  776.0f,  1072.0f, 1482.0f, 2048.0f
};

__global__ __launch_bounds__(256) void HashEncoding_kernel(
    const float* __restrict__ positions,   // [N, 3]
    const float* __restrict__ tables,      // [16, 2^19, 2]
    float*       __restrict__ out,         // [N, 32]
    int n_points)
{
    __shared__ float sp[HG_PTS_PER_BLK * 3];

    const int t        = threadIdx.x;
    const int base_pt  = blockIdx.x * HG_PTS_PER_BLK;

    // Stage this block's 16 positions (48 floats) into LDS, coalesced,
    // non-temporal (pure streaming read; keep L2 for the hash tables).
    if (t < HG_PTS_PER_BLK * 3) {
        const int gidx = base_pt * 3 + t;
        float v = 0.0f;
        if (gidx < n_points * 3)
            v = __builtin_nontemporal_load(positions + (size_t)gidx);
        sp[t] = v;
    }
    __syncthreads();

    const int pl = t >> 4;          // local point id   0..15
    const int l  = t & 15;          // level id         0..15
    const int p  = base_pt + pl;    // global point id
    if (p >= n_points) return;

    // LDS broadcast reads (16 lanes hit the same word -> no bank conflict)
    const float x = sp[pl * 3 + 0];
    const float y = sp[pl * 3 + 1];
    const float z = sp[pl * 3 + 2];

    const float res  = c_res[l];
    const float rmax = res - 1.0f;

    float fx = floorf(x * res); fx = fminf(fmaxf(fx, 0.0f), rmax);
    float fy = floorf(y * res); fy = fminf(fmaxf(fy, 0.0f), rmax);
    float fz = floorf(z * res); fz = fminf(fmaxf(fz, 0.0f), rmax);

    const uint32_t cx = (uint32_t)fx;
    const uint32_t cy = (uint32_t)fy;
    const uint32_t cz = (uint32_t)fz;

    // uint32 wraparound hash, exact mod since TABLE_SIZE | 2^32
    const uint32_t h   = cx + cy * 2654435761u + cz * 805459861u;
    const uint32_t idx = h & HG_TABLE_MASK;

    // 8-byte gather; default temporal hint -> tables stay resident in L2.
    const v2f* __restrict__ src =
        (const v2f*)tables + (((size_t)l << 19) + idx);
    const v2f feat = *src;

    // Coalesced streaming store: lanes of one 16-lane group cover one point's
    // 128 contiguous bytes -> wave emits 256B of contiguous global_store_b64.
    v2f* dst = (v2f*)out + (((size_t)p << 4) + l);
    __builtin_nontemporal_store(feat, dst);
}

extern "C" void kernel_launch(void* const* d_in, const int* in_sizes, int n_in,
                              void* d_out, int out_size, void* d_ws, size_t ws_size,
                              hipStream_t stream) {
    const float* positions = (const float*)d_in[0];   // [N, 3] float32
    const float* tables    = (const float*)d_in[1];   // [16, 524288, 2] float32
    float*       out       = (float*)d_out;           // [N, 32] float32

    const int n_points = in_sizes[0] / 3;             // 2,097,152
    const int blocks   = (n_points + HG_PTS_PER_BLK - 1) / HG_PTS_PER_BLK;

    HashEncoding_kernel<<<dim3(blocks), dim3(256), 0, stream>>>(
        positions, tables, out, n_points);
}